// ANet_74036646249366
// MI455X (gfx1250) — compile-verified
//
#include <hip/hip_runtime.h>
#include <math.h>

// ---------------- CDNA5 WMMA-based wave reduction --------------------------
// v_wmma_f32_16x16x4_f32 with B = ones:  D[m][n] = sum_k A[m][k]
// A layout (32-bit A 16x4): VGPR0 lanes0-15 -> A[m][0], lanes16-31 -> A[m][2]
// So with A = {v, 0}: D[m][n] = v_m + v_{m+16}. Summing D's 8 VGPRs gives
// lanes<16: sum rows 0..7, lanes>=16: sum rows 8..15; one xor-16 shuffle
// completes the full 32-lane sum (identical in every lane).
typedef __attribute__((ext_vector_type(2))) float v2f;
typedef __attribute__((ext_vector_type(8))) float v8f;

#define ETA_C   0.8f
#define BETA_C  0.01f
#define EPS_C   1e-6f
#define MAXIT_C 100
#define NB_RED  256   // fixed block count for deterministic partial reductions

__device__ __forceinline__ float waveReduceWMMA(float v) {
    v2f a; a[0] = v;    a[1] = 0.0f;
    v2f b; b[0] = 1.0f; b[1] = 1.0f;          // all-ones B: layout independent
    v8f c = {};
    v8f d = __builtin_amdgcn_wmma_f32_16x16x4_f32(
        /*neg_a=*/false, a, /*neg_b=*/false, b,
        /*c_mod=*/(short)0, c, /*reuse_a=*/false, /*reuse_b=*/false);
    float s = d[0] + d[1] + d[2] + d[3] + d[4] + d[5] + d[6] + d[7];
    s += __shfl_xor(s, 16, 32);               // combine the two half-wave sums
    return s;                                 // full 32-lane sum, every lane
}

__device__ __forceinline__ float blockReduceSum(float v, float* lds) {
    float w = waveReduceWMMA(v);
    __syncthreads();                          // protect lds across repeated use
    if ((threadIdx.x & 31) == 0) lds[threadIdx.x >> 5] = w;
    __syncthreads();
    float tot = 0.0f;
    if (threadIdx.x == 0) {
        int nw = blockDim.x >> 5;             // fixed-order, deterministic
        for (int i = 0; i < nw; ++i) tot += lds[i];
    }
    return tot;                               // valid in thread 0 only
}

// ---------------- kernels --------------------------------------------------

__global__ void k_init(float* __restrict__ x0, int* __restrict__ done, int n) {
    int i = blockIdx.x * blockDim.x + threadIdx.x;
    if (i < n) x0[i] = 0.0f;
    if (i == 0) *done = 0;
}

// COO SpMV scatter: y[rows[k]] += vals[k] * xin[cols[k]]  (L2-resident vector)
__global__ void k_spmv(const float* __restrict__ vals, const int* __restrict__ rows,
                       const int* __restrict__ cols, const float* __restrict__ xin,
                       float* __restrict__ y, int nnz, const int* __restrict__ done) {
    if (*done) return;
    int tid    = blockIdx.x * blockDim.x + threadIdx.x;
    int stride = gridDim.x * blockDim.x;
    int nnz4   = nnz >> 2;
    const float4* v4 = (const float4*)vals;
    const int4*   r4 = (const int4*)rows;
    const int4*   c4 = (const int4*)cols;
    for (int i = tid; i < nnz4; i += stride) {
        float4 v = v4[i]; int4 r = r4[i]; int4 c = c4[i];
        unsafeAtomicAdd(&y[r.x], v.x * xin[c.x]);
        unsafeAtomicAdd(&y[r.y], v.y * xin[c.y]);
        unsafeAtomicAdd(&y[r.z], v.z * xin[c.z]);
        unsafeAtomicAdd(&y[r.w], v.w * xin[c.w]);
    }
    for (int k = (nnz4 << 2) + tid; k < nnz; k += stride)
        unsafeAtomicAdd(&y[rows[k]], vals[k] * xin[cols[k]]);
}

// x0 = (Q@x) + x ; xi_cur = x0
__global__ void k_x0_finish(const float* __restrict__ x, float* __restrict__ x0,
                            float* __restrict__ xi, int n) {
    int i = blockIdx.x * blockDim.x + threadIdx.x;
    if (i < n) { float t = x0[i] + x[i]; x0[i] = t; xi[i] = t; }
}

// partial dot(xi,e) and dot(x0,xi)
__global__ void k_dots_partial(const float* __restrict__ xi, const float* __restrict__ e,
                               const float* __restrict__ x0, float* __restrict__ pa,
                               float* __restrict__ pb, int n, const int* __restrict__ done) {
    if (*done) return;
    __shared__ float lds[8];
    int i = blockIdx.x * blockDim.x + threadIdx.x;
    int stride = gridDim.x * blockDim.x;
    float ae = 0.0f, ax = 0.0f;
    for (; i < n; i += stride) { float t = xi[i]; ae += t * e[i]; ax += t * x0[i]; }
    float sa = blockReduceSum(ae, lds);
    float sb = blockReduceSum(ax, lds);
    if (threadIdx.x == 0) { pa[blockIdx.x] = sa; pb[blockIdx.x] = sb; }
}

__global__ void k_dots_final(const float* __restrict__ pa, const float* __restrict__ pb,
                             float* __restrict__ sc, const int* __restrict__ done) {
    if (*done) return;
    __shared__ float lds[8];
    float sa = blockReduceSum(pa[threadIdx.x], lds);
    float sb = blockReduceSum(pb[threadIdx.x], lds);
    if (threadIdx.x == 0) { sc[0] = sa; sc[1] = sb; }
}

// xi_raw = -lam*ETA*dot_e*e + (lam+BETA)*dot_x0*x0   (full overwrite)
__global__ void k_rank1(const float* __restrict__ e, const float* __restrict__ x0,
                        const float* __restrict__ sc, const float* __restrict__ lam,
                        float* __restrict__ xr, int n, const int* __restrict__ done) {
    if (*done) return;
    int i = blockIdx.x * blockDim.x + threadIdx.x;
    if (i < n) {
        float L  = lam[0];
        float c1 = -L * ETA_C * sc[0];
        float c2 = (L + BETA_C) * sc[1];
        xr[i] = c1 * e[i] + c2 * x0[i];
    }
}

__global__ void k_norm_partial(const float* __restrict__ xr, float* __restrict__ pa,
                               int n, const int* __restrict__ done) {
    if (*done) return;
    __shared__ float lds[8];
    int i = blockIdx.x * blockDim.x + threadIdx.x;
    int stride = gridDim.x * blockDim.x;
    float acc = 0.0f;
    for (; i < n; i += stride) { float t = xr[i]; acc += t * t; }
    float s = blockReduceSum(acc, lds);
    if (threadIdx.x == 0) pa[blockIdx.x] = s;
}

__global__ void k_norm_final(const float* __restrict__ pa, float* __restrict__ sc,
                             const int* __restrict__ done) {
    if (*done) return;
    __shared__ float lds[8];
    float s = blockReduceSum(pa[threadIdx.x], lds);
    if (threadIdx.x == 0) sc[2] = 1.0f / sqrtf(s);
}

// xi_new = xr * invn ; partial ||xi_new - xi||^2 ; commit xi <- xi_new.
// Runs only when done_prev == 0  => implements where(done, xi, xi_new).
__global__ void k_commit_diff(float* __restrict__ xi, const float* __restrict__ xr,
                              const float* __restrict__ sc, float* __restrict__ pa,
                              int n, const int* __restrict__ done) {
    if (*done) return;
    __shared__ float lds[8];
    int i = blockIdx.x * blockDim.x + threadIdx.x;
    int stride = gridDim.x * blockDim.x;
    float invn = sc[2];
    float acc = 0.0f;
    for (; i < n; i += stride) {
        float xn = xr[i] * invn;
        float d  = xn - xi[i];
        acc += d * d;
        xi[i] = xn;
    }
    float s = blockReduceSum(acc, lds);
    if (threadIdx.x == 0) pa[blockIdx.x] = s;
}

__global__ void k_diff_final(const float* __restrict__ pa, int* __restrict__ done) {
    if (*done) return;
    __shared__ float lds[8];
    float s = blockReduceSum(pa[threadIdx.x], lds);
    if (threadIdx.x == 0) { if (sqrtf(s) <= EPS_C) *done = 1; }
}

__global__ void k_copy(const float* __restrict__ xi, float* __restrict__ out, int n) {
    int i = blockIdx.x * blockDim.x + threadIdx.x;
    if (i < n) out[i] = xi[i];
}

// ---------------- launcher -------------------------------------------------

extern "C" void kernel_launch(void* const* d_in, const int* in_sizes, int n_in,
                              void* d_out, int out_size, void* d_ws, size_t ws_size,
                              hipStream_t stream) {
    const float* x   = (const float*)d_in[0];
    const float* e   = (const float*)d_in[1];
    const float* lam = (const float*)d_in[2];
    const float* mv  = (const float*)d_in[3];
    const float* qv  = (const float*)d_in[4];
    const int*   mr  = (const int*)d_in[5];
    const int*   mc  = (const int*)d_in[6];
    const int*   qr  = (const int*)d_in[7];
    const int*   qc  = (const int*)d_in[8];

    const int V   = in_sizes[0];
    const int NNZ = in_sizes[3];

    float* x0 = (float*)d_ws;       // V
    float* xi = x0 + V;             // V  (current iterate)
    float* xr = xi + V;             // V  (raw, un-normalized)
    float* pa = xr + V;             // NB_RED partials
    float* pb = pa + NB_RED;        // NB_RED partials
    float* sc = pb + NB_RED;        // scalars: [0]=dot_e [1]=dot_x0 [2]=invnorm
    int* done = (int*)(sc + 8);

    const int TPB = 256;
    const int gV  = (V + TPB - 1) / TPB;
    int gS = (NNZ / 4 + TPB - 1) / TPB;
    if (gS > 8192) gS = 8192;
    if (gS < 1)    gS = 1;

    // setup: x0 = Q@x + x ; xi = x0 ; done = 0
    k_init<<<gV, TPB, 0, stream>>>(x0, done, V);
    k_spmv<<<gS, TPB, 0, stream>>>(qv, qr, qc, x, x0, NNZ, done);
    k_x0_finish<<<gV, TPB, 0, stream>>>(x, x0, xi, V);

    // 1 (xi0 = F(x0), done0) + MAX_ITERS scan steps — identical bodies,
    // device-side done flag gives exact freeze semantics + early exit.
    for (int it = 0; it < 1 + MAXIT_C; ++it) {
        k_dots_partial<<<NB_RED, TPB, 0, stream>>>(xi, e, x0, pa, pb, V, done);
        k_dots_final  <<<1, NB_RED, 0, stream>>>(pa, pb, sc, done);
        k_rank1       <<<gV, TPB, 0, stream>>>(e, x0, sc, lam, xr, V, done);
        k_spmv        <<<gS, TPB, 0, stream>>>(mv, mr, mc, xi, xr, NNZ, done);
        k_norm_partial<<<NB_RED, TPB, 0, stream>>>(xr, pa, V, done);
        k_norm_final  <<<1, NB_RED, 0, stream>>>(pa, sc, done);
        k_commit_diff <<<gV, TPB, 0, stream>>>(xi, xr, sc, pa, V, done);
        k_diff_final  <<<1, NB_RED, 0, stream>>>(pa, done);
    }

    k_copy<<<gV, TPB, 0, stream>>>(xi, (float*)d_out, V);
}